// SAGE_11278584119613
// MI455X (gfx1250) — compile-verified
//
#include <hip/hip_runtime.h>
#include <hip/hip_bf16.h>

// GraphSAGE ('pool' aggregator), 3 layers on MI455X (gfx1250, wave32).
//   m      = relu(h @ Wp.T + bp)
//   pooled = segment_max(m[src], dst)   (empty -> 0; m>=0 so u32-atomic-max w/ 0-init is exact)
//   h'     = [relu](h @ Ws.T + pooled @ Wn.T + b)
// GEMMs use V_WMMA_F32_16X16X4_F32 (full fp32, f32 accumulate).

#define NNODES 50000
#define NEDGES 800000
#define HDIM   128
#define ODIM   64

typedef float v2f __attribute__((ext_vector_type(2)));
typedef float v8f __attribute__((ext_vector_type(8)));

// One wave computes a 16-row strip x (NT*16) output columns.
// A fragment (16x4 f32): lane l -> row (l&15), float2 at k + 2*(l>>4).
// B fragment (4x16 f32): lane l -> col (l&15) = W row, float2 at k + 2*(l>>4).
// C/D (16x16 f32, 8 VGPRs): vgpr v, lane l -> row v + 8*(l>>4), col (l&15).
template<int NT, bool RELU, bool DUAL>
__global__ __launch_bounds__(128) void sage_gemm_wmma(
    const float* __restrict__ X,  const float* __restrict__ W,
    const float* __restrict__ X2, const float* __restrict__ W2,
    const float* __restrict__ bias, float* __restrict__ out, int N)
{
    const int lane  = threadIdx.x & 31;
    const int wave  = threadIdx.x >> 5;
    const int r0    = (blockIdx.x * 4 + wave) * 16;   // wave-uniform strip base
    if (r0 >= N) return;                              // whole-wave exit: EXEC stays all-1s
    const int mrow  = lane & 15;
    const int khalf = (lane >> 4) * 2;                // 0 or 2

    v8f acc[NT] = {};

    {
        const float* xr = X + (size_t)(r0 + mrow) * HDIM + khalf;
        #pragma unroll 4
        for (int k = 0; k < HDIM; k += 4) {
            v2f a = *(const v2f*)(xr + k);
            #pragma unroll
            for (int t = 0; t < NT; ++t) {
                v2f bb = *(const v2f*)(W + (size_t)(t * 16 + mrow) * HDIM + khalf + k);
                acc[t] = __builtin_amdgcn_wmma_f32_16x16x4_f32(
                    false, a, false, bb, (short)0, acc[t], false, false);
            }
        }
    }
    if (DUAL) {
        const float* xr = X2 + (size_t)(r0 + mrow) * HDIM + khalf;
        #pragma unroll 4
        for (int k = 0; k < HDIM; k += 4) {
            v2f a = *(const v2f*)(xr + k);
            #pragma unroll
            for (int t = 0; t < NT; ++t) {
                v2f bb = *(const v2f*)(W2 + (size_t)(t * 16 + mrow) * HDIM + khalf + k);
                acc[t] = __builtin_amdgcn_wmma_f32_16x16x4_f32(
                    false, a, false, bb, (short)0, acc[t], false, false);
            }
        }
    }

    const int Hout = NT * 16;
    #pragma unroll
    for (int t = 0; t < NT; ++t) {
        const float bc = bias[t * 16 + mrow];
        #pragma unroll
        for (int v = 0; v < 8; ++v) {
            float val = acc[t][v] + bc;
            if (RELU) val = fmaxf(val, 0.0f);
            out[(size_t)(r0 + v + (khalf << 2)) * Hout + t * 16 + mrow] = val;
        }
    }
}

// One wave per edge; each lane handles 4 contiguous features (32*4 = 128 = HDIM).
// m >= 0 (relu), pooled initialized to 0 -> u32 atomic max == float segment_max w/ 0 fill.
__global__ __launch_bounds__(256) void sage_scatter_max(
    const float* __restrict__ m, const int* __restrict__ src,
    const int* __restrict__ dst, unsigned int* __restrict__ pooled, int E)
{
    const int tid = blockIdx.x * blockDim.x + threadIdx.x;
    const int e   = tid >> 5;
    if (e >= E) return;
    const int f = (tid & 31) * 4;
    const int s = src[e];
    const int d = dst[e];
    const float4 v = *(const float4*)(m + (size_t)s * HDIM + f);
    unsigned int* p = pooled + (size_t)d * HDIM + f;
    atomicMax(p + 0, __float_as_uint(v.x));
    atomicMax(p + 1, __float_as_uint(v.y));
    atomicMax(p + 2, __float_as_uint(v.z));
    atomicMax(p + 3, __float_as_uint(v.w));
}

__global__ __launch_bounds__(256) void sage_zero4(float4* __restrict__ p, int n4)
{
    const int i = blockIdx.x * blockDim.x + threadIdx.x;
    if (i < n4) p[i] = make_float4(0.f, 0.f, 0.f, 0.f);
}

extern "C" void kernel_launch(void* const* d_in, const int* in_sizes, int n_in,
                              void* d_out, int out_size, void* d_ws, size_t ws_size,
                              hipStream_t stream) {
    const float* x   = (const float*)d_in[0];
    const int*   src = (const int*)  d_in[1];
    const int*   dst = (const int*)  d_in[2];
    const float* Wp  = (const float*)d_in[3];
    const float* bp  = (const float*)d_in[4];
    const float* Wn  = (const float*)d_in[5];
    const float* Ws  = (const float*)d_in[6];
    const float* b   = (const float*)d_in[7];
    const float* Wp2 = (const float*)d_in[8];
    const float* bp2 = (const float*)d_in[9];
    const float* Wn2 = (const float*)d_in[10];
    const float* Ws2 = (const float*)d_in[11];
    const float* b2  = (const float*)d_in[12];
    float* out = (float*)d_out;

    const size_t NH = (size_t)NNODES * HDIM;
    float* buf0 = (float*)d_ws;       // m / h scratch
    float* buf1 = buf0 + NH;          // h scratch
    float* buf2 = buf1 + NH;          // pooled

    const int strips  = NNODES / 16;               // 3125 (exact)
    const int gblocks = (strips + 3) / 4;          // 4 waves/block
    const int zblocks = (int)((NH / 4 + 255) / 256);
    const int sblocks = (NEDGES * 32) / 256;       // 100000

    dim3 tG(128), tS(256);

    // ---- Layer 1 (input = x) ----
    sage_gemm_wmma<8, true,  false><<<gblocks, tG, 0, stream>>>(x, Wp, nullptr, nullptr, bp, buf0, NNODES);
    sage_zero4<<<zblocks, tS, 0, stream>>>((float4*)buf2, (int)(NH / 4));
    sage_scatter_max<<<sblocks, tS, 0, stream>>>(buf0, src, dst, (unsigned int*)buf2, NEDGES);
    sage_gemm_wmma<8, true,  true ><<<gblocks, tG, 0, stream>>>(x, Ws, buf2, Wn, b, buf1, NNODES);   // h1 -> buf1

    // ---- Layer 2 (shared weights, input = h1 in buf1) ----
    sage_gemm_wmma<8, true,  false><<<gblocks, tG, 0, stream>>>(buf1, Wp, nullptr, nullptr, bp, buf0, NNODES);
    sage_zero4<<<zblocks, tS, 0, stream>>>((float4*)buf2, (int)(NH / 4));
    sage_scatter_max<<<sblocks, tS, 0, stream>>>(buf0, src, dst, (unsigned int*)buf2, NEDGES);
    sage_gemm_wmma<8, true,  true ><<<gblocks, tG, 0, stream>>>(buf1, Ws, buf2, Wn, b, buf0, NNODES); // h2 -> buf0

    // ---- Layer 3 (final, H -> O, no relu) ----
    sage_gemm_wmma<8, true,  false><<<gblocks, tG, 0, stream>>>(buf0, Wp2, nullptr, nullptr, bp2, buf1, NNODES);
    sage_zero4<<<zblocks, tS, 0, stream>>>((float4*)buf2, (int)(NH / 4));
    sage_scatter_max<<<sblocks, tS, 0, stream>>>(buf1, src, dst, (unsigned int*)buf2, NEDGES);
    sage_gemm_wmma<4, false, true ><<<gblocks, tG, 0, stream>>>(buf0, Ws2, buf2, Wn2, b2, out, NNODES); // [N,64]
}